// AttnDecoder_47493748359495
// MI455X (gfx1250) — compile-verified
//
#include <hip/hip_runtime.h>

typedef __attribute__((ext_vector_type(2))) float v2f;
typedef __attribute__((ext_vector_type(4))) float v4f;
typedef __attribute__((ext_vector_type(8))) float v8f;
typedef __attribute__((ext_vector_type(4))) int   v4i;

#define BB 64
#define LL 256
#define EE 512
#define HH 1024
#define VV 32000
#define NEGBIG (-2147483647.0f)

// ---- optional CDNA5 async global->LDS path (probe-guarded) -----------------
#if __has_builtin(__builtin_amdgcn_global_load_async_to_lds_b128) && \
    __has_builtin(__builtin_amdgcn_s_wait_asynccnt)
#define HAVE_ASYNC_LDS 1
// Param 1 (per hipcc diagnostic): int4 in __device__ (addrspace(1)); assume
// param 2 is the symmetric int4 in __shared__ (addrspace(3)).
typedef __attribute__((address_space(1))) v4i as1_v4i;
typedef __attribute__((address_space(3))) v4i as3_v4i;
#define TO_AS1(p) ((as1_v4i*)(uintptr_t)(p))
#define TO_AS3(p) ((as3_v4i*)(unsigned)(uintptr_t)(p))
#endif

#define KC      64    // K-chunk staged in LDS
#define APITCH  68    // 64+4  : row stride ≡ 4 (mod 64 banks) -> conflict-free A frags
#define WPITCH 144    // 128+16: 2*stride ≡ 32 (mod 64) -> lane halves on disjoint banks

// ---------------------------------------------------------------------------
// C[64 x N] = A[64 x K] @ W[K x N] (+bias), fp32 WMMA 16x16x4.
// Block = 8 waves = 128 columns, all 64 rows. A and W chunks are staged in
// LDS with coalesced b128 (async-to-LDS when available); each wave owns one
// 16-col strip and 4 row tiles, reading fragments from LDS.
// ---------------------------------------------------------------------------
__global__ __launch_bounds__(256) void gemm64_wmma_f32(
    const float* __restrict__ A, const float* __restrict__ W,
    const float* __restrict__ bias, float* __restrict__ C,
    int K, int N)
{
  __shared__ float sA[64 * APITCH];   // 17.4 KB
  __shared__ float sW[KC * WPITCH];   // 36.9 KB

  const int tid  = threadIdx.x;
  const int lane = tid & 31;
  const int wave = tid >> 5;
  const int cb   = blockIdx.x * 128;
  const int col0 = cb + wave * 16;

  const int r  = lane & 15;           // row-in-tile / column index
  const int kh = (lane >> 4) << 1;    // k sub-offset per lane half
  const int mb = (lane >> 4) << 3;    // C/D: VGPR i -> M=i / M=i+8

  v8f acc0 = {}, acc1 = {}, acc2 = {}, acc3 = {};

  const float* a0p = &sA[(size_t)(r)      * APITCH + kh];
  const float* a1p = &sA[(size_t)(16 + r) * APITCH + kh];
  const float* a2p = &sA[(size_t)(32 + r) * APITCH + kh];
  const float* a3p = &sA[(size_t)(48 + r) * APITCH + kh];
  const float* wfp = &sW[wave * 16 + r];

  for (int k0 = 0; k0 < K; k0 += KC) {
    __syncthreads();                  // previous chunk fully consumed

    // ---- stage A chunk: 64 rows x KC, coalesced b128 ----
    #pragma unroll
    for (int i = 0; i < (64 * KC) / (256 * 4); ++i) {      // 4
      const int f   = i * 256 + tid;                        // float4 id
      const int row = f >> 4;                               // 16 float4 / row
      const int c4  = (f & 15) << 2;
      const float* gp = A + (size_t)row * K + k0 + c4;
      float* lp = &sA[row * APITCH + c4];
#ifdef HAVE_ASYNC_LDS
      __builtin_amdgcn_global_load_async_to_lds_b128(TO_AS1(gp), TO_AS3(lp), 0, 0);
#else
      *(v4f*)lp = *(const v4f*)gp;
#endif
    }
    // ---- stage W chunk: KC rows x 128 cols, coalesced b128 ----
    #pragma unroll
    for (int i = 0; i < (KC * 128) / (256 * 4); ++i) {     // 8
      const int f   = i * 256 + tid;
      const int row = f >> 5;                               // 32 float4 / row
      const int c4  = (f & 31) << 2;
      const float* gp = W + (size_t)(k0 + row) * N + cb + c4;
      float* lp = &sW[row * WPITCH + c4];
#ifdef HAVE_ASYNC_LDS
      __builtin_amdgcn_global_load_async_to_lds_b128(TO_AS1(gp), TO_AS3(lp), 0, 0);
#else
      *(v4f*)lp = *(const v4f*)gp;
#endif
    }
#ifdef HAVE_ASYNC_LDS
    __builtin_amdgcn_s_wait_asynccnt(0);
#endif
    __syncthreads();

    // ---- compute the chunk from LDS ----
    #pragma unroll 8
    for (int kk = 0; kk < KC; kk += 4) {
      v2f a0 = *(const v2f*)(a0p + kk);
      v2f a1 = *(const v2f*)(a1p + kk);
      v2f a2 = *(const v2f*)(a2p + kk);
      v2f a3 = *(const v2f*)(a3p + kk);
      v2f b;
      b.x = wfp[(kk + kh)     * WPITCH];
      b.y = wfp[(kk + kh + 1) * WPITCH];
      acc0 = __builtin_amdgcn_wmma_f32_16x16x4_f32(false, a0, false, b, (short)0, acc0, false, false);
      acc1 = __builtin_amdgcn_wmma_f32_16x16x4_f32(false, a1, false, b, (short)0, acc1, false, false);
      acc2 = __builtin_amdgcn_wmma_f32_16x16x4_f32(false, a2, false, b, (short)0, acc2, false, false);
      acc3 = __builtin_amdgcn_wmma_f32_16x16x4_f32(false, a3, false, b, (short)0, acc3, false, false);
    }
  }

  const float bv = bias ? bias[col0 + r] : 0.0f;
  #pragma unroll
  for (int i = 0; i < 8; ++i) {
    const int m = mb + i;
    C[(size_t)(m)      * N + col0 + r] = acc0[i] + bv;
    C[(size_t)(16 + m) * N + col0 + r] = acc1[i] + bv;
    C[(size_t)(32 + m) * N + col0 + r] = acc2[i] + bv;
    C[(size_t)(48 + m) * N + col0 + r] = acc3[i] + bv;
  }
}

// flat[b, i*H + h] = hidden[i, b, h]   (64 x 2048)
__global__ void pack_flat_k(const float* __restrict__ hidden, float* __restrict__ flat) {
  int idx = blockIdx.x * blockDim.x + threadIdx.x;
  int b = idx >> 11, j = idx & 2047;
  int layer = j >> 10, h = j & 1023;
  flat[idx] = hidden[((size_t)layer * BB + b) * HH + h];
}

// ---------------------------------------------------------------------------
// Fused masked softmax-attention (online/flash style): single pass over
// q_output computes scores, softmax, and context — halves q traffic vs the
// two-pass formulation (67 MB instead of 134 MB of HBM reads).
// ---------------------------------------------------------------------------
__global__ __launch_bounds__(256) void attn_fused_k(
    const float* __restrict__ e, const float* __restrict__ q,
    const unsigned char* __restrict__ qmask, float* __restrict__ ctx)
{
  const int b = blockIdx.x;
  const int lane = threadIdx.x & 31, wave = threadIdx.x >> 5;
  __shared__ float s_m[8], s_s[8];
  __shared__ float s_acc[8][HH];   // 32 KB

  const float* eb = e + (size_t)b * HH;
  float er[32];
  #pragma unroll
  for (int j = 0; j < 32; ++j) er[j] = eb[j * 32 + lane];

  float m = -3.0e38f, s = 0.0f;
  float acc[32];
  #pragma unroll
  for (int j = 0; j < 32; ++j) acc[j] = 0.0f;

  for (int l = wave; l < LL; l += 8) {
    const float* qrow = q + ((size_t)b * LL + l) * HH;
    float qr[32];
    float d = 0.0f;
    #pragma unroll
    for (int j = 0; j < 32; ++j) { qr[j] = qrow[j * 32 + lane]; d += er[j] * qr[j]; }
    #pragma unroll
    for (int off = 16; off > 0; off >>= 1) d += __shfl_xor(d, off, 32);
    const float sl = qmask[(size_t)b * LL + l] ? d : NEGBIG;
    const float nm = fmaxf(m, sl);
    const float scale = __expf(m - nm);
    const float w = __expf(sl - nm);
    s = s * scale + w;
    #pragma unroll
    for (int j = 0; j < 32; ++j) acc[j] = acc[j] * scale + w * qr[j];
    m = nm;
  }
  #pragma unroll
  for (int j = 0; j < 32; ++j) s_acc[wave][j * 32 + lane] = acc[j];
  if (lane == 0) { s_m[wave] = m; s_s[wave] = s; }
  __syncthreads();

  float M = -3.0e38f;
  #pragma unroll
  for (int w2 = 0; w2 < 8; ++w2) M = fmaxf(M, s_m[w2]);
  float S = 0.0f, fw[8];
  #pragma unroll
  for (int w2 = 0; w2 < 8; ++w2) { fw[w2] = __expf(s_m[w2] - M); S += s_s[w2] * fw[w2]; }
  const float inv = 1.0f / S;
  for (int h = threadIdx.x; h < HH; h += 256) {
    float v = 0.0f;
    #pragma unroll
    for (int w2 = 0; w2 < 8; ++w2) v += fw[w2] * s_acc[w2][h];
    ctx[(size_t)b * HH + h] = v * inv;
  }
}

// x0[b, :] = concat(a_embed[b,0,:], ctx[b,:])    (64 x 1536)
__global__ void pack_x0_k(const float* __restrict__ ae, const float* __restrict__ ctx,
                          float* __restrict__ x0) {
  int idx = blockIdx.x * blockDim.x + threadIdx.x;
  int b = idx / (EE + HH), j = idx % (EE + HH);
  x0[idx] = (j < EE) ? ae[(size_t)b * EE + j] : ctx[(size_t)b * HH + (j - EE)];
}

// GRU gate elementwise (Keras reset_after)
__global__ void gru_elem_k(const float* __restrict__ mx, const float* __restrict__ mi,
                           const float* __restrict__ hprev, const unsigned char* __restrict__ amask,
                           float* __restrict__ state_out, float* __restrict__ x_next)
{
  const int idx = blockIdx.x * blockDim.x + threadIdx.x;
  const int b = idx >> 10, h = idx & 1023;
  const size_t o = (size_t)b * 3 * HH;
  const float xz = mx[o + h], xr = mx[o + HH + h], xh = mx[o + 2 * HH + h];
  const float rz = mi[o + h], rr = mi[o + HH + h], rh = mi[o + 2 * HH + h];
  const float z  = 1.0f / (1.0f + __expf(-(xz + rz)));
  const float r  = 1.0f / (1.0f + __expf(-(xr + rr)));
  const float hh = tanhf(xh + r * rh);
  const float hold = hprev[idx];
  float hn = z * hold + (1.0f - z) * hh;
  const bool mm = amask[b] != 0;
  hn = mm ? hn : hold;
  state_out[idx] = hn;
  x_next[idx]    = mm ? hn : 0.0f;
}

extern "C" void kernel_launch(void* const* d_in, const int* in_sizes, int n_in,
                              void* d_out, int out_size, void* d_ws, size_t ws_size,
                              hipStream_t stream) {
  const float* a_embed = (const float*)d_in[0];
  const float* q_out   = (const float*)d_in[1];
  const float* hidden  = (const float*)d_in[2];
  const unsigned char* q_mask = (const unsigned char*)d_in[3];
  const unsigned char* a_mask = (const unsigned char*)d_in[4];
  const float* W_attn = (const float*)d_in[5];
  const float* K0  = (const float*)d_in[6];
  const float* R0  = (const float*)d_in[7];
  const float* bi0 = (const float*)d_in[8];
  const float* br0 = (const float*)d_in[9];
  const float* K1  = (const float*)d_in[10];
  const float* R1  = (const float*)d_in[11];
  const float* bi1 = (const float*)d_in[12];
  const float* br1 = (const float*)d_in[13];
  const float* Wp  = (const float*)d_in[14];
  const float* bp  = (const float*)d_in[15];

  float* out = (float*)d_out;
  float* a_output  = out;                       // 64*32000
  float* state_out = out + (size_t)BB * VV;     // 2*64*1024

  float* ws   = (float*)d_ws;
  float* flat = ws;                  // 64*2048 = 131072
  float* e    = flat + 131072;       // 64*1024 =  65536
  float* ctx  = e    + 65536;        // 64*1024 =  65536
  float* x0   = ctx  + 65536;        // 64*1536 =  98304
  float* mx   = x0   + 98304;        // 64*3072 = 196608
  float* mi   = mx   + 196608;       // 64*3072 = 196608
  float* x1   = mi   + 196608;       // 64*1024 =  65536
  float* x2   = x1   + 65536;        // 64*1024 =  65536

  pack_flat_k<<<(BB * 2 * HH) / 256, 256, 0, stream>>>(hidden, flat);
  gemm64_wmma_f32<<<HH / 128, 256, 0, stream>>>(flat, W_attn, nullptr, e, 2 * HH, HH);
  attn_fused_k<<<BB, 256, 0, stream>>>(e, q_out, q_mask, ctx);
  pack_x0_k<<<(BB * (EE + HH)) / 256, 256, 0, stream>>>(a_embed, ctx, x0);

  gemm64_wmma_f32<<<3 * HH / 128, 256, 0, stream>>>(x0, K0, bi0, mx, EE + HH, 3 * HH);
  gemm64_wmma_f32<<<3 * HH / 128, 256, 0, stream>>>(hidden, R0, br0, mi, HH, 3 * HH);
  gru_elem_k<<<(BB * HH) / 256, 256, 0, stream>>>(mx, mi, hidden, a_mask, state_out, x1);

  gemm64_wmma_f32<<<3 * HH / 128, 256, 0, stream>>>(x1, K1, bi1, mx, HH, 3 * HH);
  gemm64_wmma_f32<<<3 * HH / 128, 256, 0, stream>>>(hidden + (size_t)BB * HH, R1, br1, mi, HH, 3 * HH);
  gru_elem_k<<<(BB * HH) / 256, 256, 0, stream>>>(mx, mi, hidden + (size_t)BB * HH, a_mask,
                                                  state_out + (size_t)BB * HH, x2);

  gemm64_wmma_f32<<<VV / 128, 256, 0, stream>>>(x2, Wp, bp, a_output, HH, VV);
}